// GeneralizedRCNNTransform_44435731644934
// MI455X (gfx1250) — compile-verified
//
#include <hip/hip_runtime.h>
#include <hip/hip_bf16.h>

// GeneralizedRCNNTransform: normalize + bilinear resize (600x900 -> 799x1199,
// half-pixel centers, align_corners=False) + zero-pad to (800,1216).
// Memory-bound (~145 MB @ 23.3 TB/s ~ 6.3 us). Data movement via CDNA5 TDM:
// one tensor_load_to_lds per workgroup stages the needed input row slab; the
// bilinear gathers then hit LDS instead of re-reading L2/HBM ~4x. Output is
// streamed with 128-bit non-temporal stores (4 px per lane).

#define IN_H   600
#define IN_W   900
#define OUT_HV 799   // valid (resized) rows
#define OUT_WV 1199  // valid (resized) cols
#define OUT_H  800   // padded rows
#define OUT_W  1216  // padded cols
#define TILE_H 16
#define ROWS_MAX 16  // worst case input rows per tile: floor(15*600/799)+3 = 14

typedef unsigned int v4u __attribute__((ext_vector_type(4)));
typedef int          v4i __attribute__((ext_vector_type(4)));
typedef int          v8i __attribute__((ext_vector_type(8)));
typedef float        f4  __attribute__((ext_vector_type(4)));

__global__ __launch_bounds__(256) void
rcnn_transform_kernel(const float* __restrict__ img,
                      const float* __restrict__ mean,
                      const float* __restrict__ stdv,
                      float* __restrict__ out) {
    __shared__ __align__(128) float smem[ROWS_MAX * IN_W];

    const int c   = blockIdx.y;
    const int b   = blockIdx.z;
    const int oy0 = blockIdx.x * TILE_H;

    const float RY = (float)IN_H / (float)OUT_HV;   // 600/799
    const float RX = (float)IN_W / (float)OUT_WV;   // 900/1199

    // Input row range needed by this tile's valid output rows.
    int oy_lastv = oy0 + TILE_H - 1;
    if (oy_lastv > OUT_HV - 1) oy_lastv = OUT_HV - 1;
    int y_start = (int)floorf(((float)oy0 + 0.5f) * RY - 0.5f);
    if (y_start < 0) y_start = 0;
    int y_end = (int)floorf(((float)oy_lastv + 0.5f) * RY - 0.5f) + 1;
    if (y_end > IN_H - 1) y_end = IN_H - 1;
    int nrows = y_end - y_start + 1;
    if (nrows > ROWS_MAX) nrows = ROWS_MAX;

    // --- TDM: DMA an nrows x IN_W fp32 slab (row stride IN_W) into LDS. ---
    // Issued by wave 0 only (other waves branch around via execz). Wave 0
    // drains TENSORcnt; the barrier then publishes the LDS tile to all waves.
    if (threadIdx.x < 32) {
        unsigned long long ga = (unsigned long long)(const void*)
            (img + ((size_t)(b * 3 + c) * IN_H + (size_t)y_start) * IN_W);
        unsigned int lds_off = (unsigned int)(unsigned long long)(const void*)&smem[0];

        // D# group 0: count=1 | lds_addr | global_addr[56:0] | type=2
        v4u g0;
        g0.x = 1u;                                   // count=1, user mode
        g0.y = lds_off;                              // LDS byte address
        g0.z = (unsigned int)(ga & 0xFFFFFFFFull);   // global_addr[31:0]
        g0.w = (unsigned int)((ga >> 32) & 0x1FFFFFFull) | (2u << 30); // [56:32], type=2

        // D# group 1: data_size=4B; tensor_dim0=900; tensor_dim1=nrows;
        // tile_dim0=900; tile_dim1=nrows; tensor_dim0_stride=900.
        unsigned int nr = (unsigned int)nrows & 0xFFFFu;
        v8i g1;
        g1[0] = (int)(2u << 16);                     // data_size=2 (4 bytes)
        g1[1] = (int)((unsigned)IN_W << 16);         // tensor_dim0[15:0] @bits63:48
        g1[2] = (int)(nr << 16);                     // tensor_dim1[15:0] @bits95:80
        g1[3] = (int)((unsigned)IN_W << 16);         // tile_dim0 @bits127:112
        g1[4] = (int)nr;                             // tile_dim1 @bits143:128 (tile_dim2=0)
        g1[5] = (int)IN_W;                           // tensor_dim0_stride[31:0]
        g1[6] = 0;                                   // stride hi / dim1_stride lo
        g1[7] = 0;

        v4i gz = {0, 0, 0, 0};                       // groups 2/3: unused (2D)
#if defined(__clang_major__) && __clang_major__ >= 23
        v8i gz8 = {0, 0, 0, 0, 0, 0, 0, 0};
        __builtin_amdgcn_tensor_load_to_lds(g0, g1, gz, gz, gz8, 0);
#else
        __builtin_amdgcn_tensor_load_to_lds(g0, g1, gz, gz, 0);
#endif
        __builtin_amdgcn_s_wait_tensorcnt(0);
    }
    __syncthreads();

    const float m      = mean[c];
    const float inv_sd = 1.0f / stdv[c];
    const size_t plane = ((size_t)(b * 3 + c)) * OUT_H * OUT_W;

    // Coalesced output: lanes stride across columns, 4 px (b128) per lane.
    for (int r = 0; r < TILE_H; ++r) {
        const int oy = oy0 + r;
        float* orow = &out[plane + (size_t)oy * OUT_W];

        if (oy >= OUT_HV) {
            // Bottom pad row (only in the last tile): stream zeros.
            const f4 z = {0.0f, 0.0f, 0.0f, 0.0f};
            for (int xb = (int)threadIdx.x * 4; xb < OUT_W; xb += (int)blockDim.x * 4)
                __builtin_nontemporal_store(z, (f4*)&orow[xb]);
            continue;
        }

        float sy = ((float)oy + 0.5f) * RY - 0.5f;
        int   y0 = (int)floorf(sy);
        float fy = sy - (float)y0;
        int y0c = y0 < 0 ? 0 : y0;
        int y1c = y0 + 1 > IN_H - 1 ? IN_H - 1 : y0 + 1;
        const float* row0 = &smem[(y0c - y_start) * IN_W];
        const float* row1 = &smem[(y1c - y_start) * IN_W];

        for (int xb = (int)threadIdx.x * 4; xb < OUT_W; xb += (int)blockDim.x * 4) {
            f4 v;
#pragma unroll
            for (int j = 0; j < 4; ++j) {
                const int ox = xb + j;
                float e = 0.0f;
                if (ox < OUT_WV) {
                    float sx = ((float)ox + 0.5f) * RX - 0.5f;
                    int   x0 = (int)floorf(sx);
                    float fx = sx - (float)x0;
                    int x0c = x0 < 0 ? 0 : x0;
                    int x1c = x0 + 1 > IN_W - 1 ? IN_W - 1 : x0 + 1;

                    float a00 = row0[x0c], a01 = row0[x1c];
                    float a10 = row1[x0c], a11 = row1[x1c];
                    float top = fmaf(fx, a01 - a00, a00);
                    float bot = fmaf(fx, a11 - a10, a10);
                    float val = fmaf(fy, bot - top, top);
                    e = (val - m) * inv_sd;
                }
                v[j] = e;
            }
            // Streaming 128-bit store (output never re-read; right pad = 0).
            __builtin_nontemporal_store(v, (f4*)&orow[xb]);
        }
    }
}

extern "C" void kernel_launch(void* const* d_in, const int* in_sizes, int n_in,
                              void* d_out, int out_size, void* d_ws, size_t ws_size,
                              hipStream_t stream) {
    const float* img  = (const float*)d_in[0];  // (8,3,600,900) fp32
    const float* mean = (const float*)d_in[1];  // (3,)
    const float* stdv = (const float*)d_in[2];  // (3,)
    float* out = (float*)d_out;                 // (8,3,800,1216) fp32

    dim3 grid(OUT_H / TILE_H, 3, 8);            // 50 x 3 x 8
    dim3 block(256);
    rcnn_transform_kernel<<<grid, block, 0, stream>>>(img, mean, stdv, out);
}